// ResidualBlock_30614526886066
// MI455X (gfx1250) — compile-verified
//
#include <hip/hip_runtime.h>
#include <hip/hip_bf16.h>

typedef float v2f __attribute__((ext_vector_type(2)));
typedef float v8f __attribute__((ext_vector_type(8)));

#define DD 128            // feature dim
#define LN_EPS 1e-5f

// ---------------------------------------------------------------------------
// 1) degree accumulation: deg[dst] += 1  (deg buffer pre-zeroed)
// ---------------------------------------------------------------------------
__global__ __launch_bounds__(256) void deg_kernel(const long long* __restrict__ dst,
                                                  float* __restrict__ deg, int E) {
    int e = blockIdx.x * blockDim.x + threadIdx.x;
    if (e < E) atomicAdd(&deg[(int)dst[e]], 1.0f);
}

// deg -> dinv = rsqrt(deg + 1)   (the +1 is the self loop)
__global__ __launch_bounds__(256) void dinv_kernel(float* __restrict__ d, int N) {
    int i = blockIdx.x * blockDim.x + threadIdx.x;
    if (i < N) d[i] = rsqrtf(d[i] + 1.0f);
}

// ---------------------------------------------------------------------------
// 2) C[N x 128] = A[N x 128] * W[128 x 128], fp32 WMMA (exact precision).
//    Block = 256 threads = 8 waves. Block handles a 16-row strip; wave w owns
//    the 16x16 output tile at columns [16w, 16w+16).
//    A strip staged in LDS with row stride 132 floats so the K-strided
//    A-fragment reads hit distinct banks ((m*132+k)%64 = (4m+k)%64).
//    Fragment layouts per CDNA5 ISA 7.12.2:
//      A 16x4 f32 : VGPR0 = {K=k0 | K=k0+2}, VGPR1 = {K=k0+1 | K=k0+3},
//                   M = lane%16, halves split on lane>=16
//      B 4x16 f32 : rows striped across lanes, same half/K split
//      C/D 16x16  : VGPR v -> M = v + 8*(lane>=16), N = lane%16
// ---------------------------------------------------------------------------
__global__ __launch_bounds__(256) void gemm_xw_wmma(const float* __restrict__ A,
                                                    const float* __restrict__ W,
                                                    float* __restrict__ C, int N) {
    __shared__ float As[16 * 132];

    const int tid  = threadIdx.x;
    const int wave = tid >> 5;
    const int lane = tid & 31;
    const int rowbase = blockIdx.x * 16;

    // Stage 16x128 A strip (coalesced: consecutive threads -> consecutive k)
    #pragma unroll
    for (int i = 0; i < 8; ++i) {
        int idx = tid + i * 256;           // 0..2047
        int m = idx >> 7, k = idx & 127;
        As[m * 132 + k] = A[(size_t)(rowbase + m) * DD + k];
    }
    __syncthreads();

    const int half = lane >> 4;            // 0: lanes 0-15, 1: lanes 16-31
    const int l15  = lane & 15;
    const int colbase = wave * 16;

    v8f acc = {};
    #pragma unroll 4
    for (int k0 = 0; k0 < DD; k0 += 4) {
        const int ka = k0 + 2 * half;
        v2f a, b;
        a.x = As[l15 * 132 + ka];
        a.y = As[l15 * 132 + ka + 1];
        // B from global: 64B-contiguous per half-wave, W stays hot in cache
        b.x = W[(size_t)ka * DD + colbase + l15];
        b.y = W[(size_t)(ka + 1) * DD + colbase + l15];
        acc = __builtin_amdgcn_wmma_f32_16x16x4_f32(
            /*neg_a=*/false, a, /*neg_b=*/false, b,
            /*c_mod=*/(short)0, acc, /*reuse_a=*/false, /*reuse_b=*/false);
    }

    const int row0 = rowbase + half * 8;
    const int col  = colbase + l15;
    #pragma unroll
    for (int v = 0; v < 8; ++v)
        C[(size_t)(row0 + v) * DD + col] = acc[v];
}

// ---------------------------------------------------------------------------
// 3) edge scatter: one wave32 per edge, float4 per lane (4*32 = 128 feats).
//    agg[dst] += h[src] * dinv[src]*dinv[dst]   via L2-resident f32 atomics.
// ---------------------------------------------------------------------------
__global__ __launch_bounds__(256) void edge_scatter(const float* __restrict__ h,
                                                    const long long* __restrict__ src,
                                                    const long long* __restrict__ dst,
                                                    const float* __restrict__ dinv,
                                                    float* __restrict__ agg, int E) {
    int gid  = blockIdx.x * blockDim.x + threadIdx.x;
    int e    = gid >> 5;
    int lane = gid & 31;
    if (e >= E) return;
    const int s = (int)src[e];
    const int d = (int)dst[e];
    const float norm = dinv[s] * dinv[d];
    const float4 hv = ((const float4*)(h + (size_t)s * DD))[lane];
    float* out = agg + (size_t)d * DD + lane * 4;
    atomicAdd(out + 0, hv.x * norm);
    atomicAdd(out + 1, hv.y * norm);
    atomicAdd(out + 2, hv.z * norm);
    atomicAdd(out + 3, hv.w * norm);
}

// ---------------------------------------------------------------------------
// 4) fused: v = agg + h*dinv^2 + bias ; LayerNorm ; ReLU ; (+resid) -> out
//    One wave32 per node, 4 floats per lane, shfl_xor butterfly reductions.
// ---------------------------------------------------------------------------
__global__ __launch_bounds__(256) void node_finalize(const float* __restrict__ agg,
                                                     const float* __restrict__ h,
                                                     const float* __restrict__ dinv,
                                                     const float* __restrict__ bias,
                                                     const float* __restrict__ ln_g,
                                                     const float* __restrict__ ln_b,
                                                     const float* __restrict__ resid, // may be null
                                                     float* __restrict__ out, int N) {
    const int wave = threadIdx.x >> 5;
    const int lane = threadIdx.x & 31;
    const int i = blockIdx.x * 8 + wave;
    if (i >= N) return;

    const float dv = dinv[i];
    const float selfw = dv * dv;

    const float4 av = ((const float4*)(agg + (size_t)i * DD))[lane];
    const float4 hv = ((const float4*)(h   + (size_t)i * DD))[lane];
    const float4 bv = ((const float4*)bias)[lane];

    float4 v;
    v.x = av.x + hv.x * selfw + bv.x;
    v.y = av.y + hv.y * selfw + bv.y;
    v.z = av.z + hv.z * selfw + bv.z;
    v.w = av.w + hv.w * selfw + bv.w;

    // mean
    float s = v.x + v.y + v.z + v.w;
    #pragma unroll
    for (int off = 16; off > 0; off >>= 1) s += __shfl_xor(s, off, 32);
    const float mean = s * (1.0f / DD);

    float4 c;
    c.x = v.x - mean; c.y = v.y - mean; c.z = v.z - mean; c.w = v.w - mean;

    // variance
    float q = c.x * c.x + c.y * c.y + c.z * c.z + c.w * c.w;
    #pragma unroll
    for (int off = 16; off > 0; off >>= 1) q += __shfl_xor(q, off, 32);
    const float r = rsqrtf(q * (1.0f / DD) + LN_EPS);

    const float4 gv  = ((const float4*)ln_g)[lane];
    const float4 bbv = ((const float4*)ln_b)[lane];

    float4 y;
    y.x = fmaxf(c.x * r * gv.x + bbv.x, 0.0f);
    y.y = fmaxf(c.y * r * gv.y + bbv.y, 0.0f);
    y.z = fmaxf(c.z * r * gv.z + bbv.z, 0.0f);
    y.w = fmaxf(c.w * r * gv.w + bbv.w, 0.0f);

    if (resid) {
        const float4 xv = ((const float4*)(resid + (size_t)i * DD))[lane];
        y.x += xv.x; y.y += xv.y; y.z += xv.z; y.w += xv.w;
    }
    ((float4*)(out + (size_t)i * DD))[lane] = y;
}

// ---------------------------------------------------------------------------
extern "C" void kernel_launch(void* const* d_in, const int* in_sizes, int n_in,
                              void* d_out, int out_size, void* d_ws, size_t ws_size,
                              hipStream_t stream) {
    const float*     x    = (const float*)d_in[0];
    const long long* eidx = (const long long*)d_in[1];   // int64 [2, E]
    const float*     W1   = (const float*)d_in[2];
    const float*     b1   = (const float*)d_in[3];
    const float*     g1   = (const float*)d_in[4];
    const float*     bb1  = (const float*)d_in[5];
    const float*     W2   = (const float*)d_in[6];
    const float*     b2   = (const float*)d_in[7];
    const float*     g2   = (const float*)d_in[8];
    const float*     bb2  = (const float*)d_in[9];
    float*           out  = (float*)d_out;

    const int N = in_sizes[0] / DD;       // 50000
    const int E = in_sizes[1] / 2;        // 600000
    const long long* src = eidx;
    const long long* dst = eidx + E;

    // workspace layout (floats): dinv[N] | bufH[N*128] | bufA[N*128] | bufB[N*128]
    float* dinv = (float*)d_ws;
    float* bufH = dinv + N;
    float* bufA = bufH + (size_t)N * DD;
    float* bufB = bufA + (size_t)N * DD;

    const size_t featBytes = (size_t)N * DD * sizeof(float);

    // degrees -> dinv
    hipMemsetAsync(dinv, 0, (size_t)N * sizeof(float), stream);
    deg_kernel<<<(E + 255) / 256, 256, 0, stream>>>(dst, dinv, E);
    dinv_kernel<<<(N + 255) / 256, 256, 0, stream>>>(dinv, N);

    // ---- layer 1 ----
    gemm_xw_wmma<<<N / 16, 256, 0, stream>>>(x, W1, bufH, N);
    hipMemsetAsync(bufA, 0, featBytes, stream);
    edge_scatter<<<(E * 32 + 255) / 256, 256, 0, stream>>>(bufH, src, dst, dinv, bufA, E);
    node_finalize<<<(N + 7) / 8, 256, 0, stream>>>(bufA, bufH, dinv, b1, g1, bb1,
                                                   nullptr, bufA, N);

    // ---- layer 2 ----
    gemm_xw_wmma<<<N / 16, 256, 0, stream>>>(bufA, W2, bufH, N);
    hipMemsetAsync(bufB, 0, featBytes, stream);
    edge_scatter<<<(E * 32 + 255) / 256, 256, 0, stream>>>(bufH, src, dst, dinv, bufB, E);
    node_finalize<<<(N + 7) / 8, 256, 0, stream>>>(bufB, bufH, dinv, b2, g2, bb2,
                                                   x, out, N);
}